// WeightTransformKernelBilateralFilter_75926431858849
// MI455X (gfx1250) — compile-verified
//
#include <hip/hip_runtime.h>
#include <cstdint>

#define B_   2
#define C_   8
#define H_   1088
#define W_   1920
#define TILE 16
#define HALO 4                    // r * dilation = 2 * 2
#define LW   (TILE + 2*HALO)      // 24
#define LPIX (LW * LW)            // 576 tile pixels
#define LELEMS (C_ * LPIX)        // 4608 floats  (18.4 KB LDS)
#define LVEC4  (LELEMS / 4)       // 1152 float4 chunks

// ---------------------------------------------------------------------------
// Kernel 1: weight transform (tiny, one thread does everything)
//   wsf[0..7]  = -params[c]^2 * log2(e)          (range coeffs, exp2-folded)
//   wsf[8..32] = log2(tk[ky][kx]) + (sx*dx^2 + sy*dy^2)*log2(e)   per tap
// ---------------------------------------------------------------------------
__global__ void bf_weight_xform(const float* __restrict__ params,
                                const float* __restrict__ kern,
                                const int*   __restrict__ dil_p,
                                float*       __restrict__ wsf) {
  if (threadIdx.x != 0 || blockIdx.x != 0) return;
  const float LOG2E = 1.4426950408889634f;
  float tp[C_ + 2];
  #pragma unroll
  for (int i = 0; i < C_ + 2; ++i) { float p = params[i]; tp[i] = -p * p; }
  #pragma unroll
  for (int c = 0; c < C_; ++c) wsf[c] = tp[c] * LOG2E;

  // softmax(kernel / 5) in log2 space
  float mx = -3.4e38f;
  for (int i = 0; i < 25; ++i) { float v = kern[i] * 0.2f; mx = v > mx ? v : mx; }
  float s = 0.f;
  for (int i = 0; i < 25; ++i) s += __expf(kern[i] * 0.2f - mx);
  const float l2s = __log2f(s);

  const int dil = *dil_p;
  for (int ky = 0; ky < 5; ++ky) {
    for (int kx = 0; kx < 5; ++kx) {
      const int i = ky * 5 + kx;
      const float dy = (float)((ky - 2) * dil);
      const float dx = (float)((kx - 2) * dil);
      const float ltk = (kern[i] * 0.2f - mx) * LOG2E - l2s;       // log2(tk_i)
      wsf[C_ + i] = ltk + (tp[C_] * dx * dx + tp[C_ + 1] * dy * dy) * LOG2E;
    }
  }
}

// ---------------------------------------------------------------------------
// gfx1250 async global->LDS copy (ISA ch10.7 / 15.18, tracked by ASYNCcnt)
// ---------------------------------------------------------------------------
__device__ __forceinline__ void async_g2l_b128(unsigned lds_byte_off, const float* g) {
  asm volatile("global_load_async_to_lds_b128 %0, %1, off"
               :: "v"(lds_byte_off), "v"(g)
               : "memory");
}
__device__ __forceinline__ void wait_asynccnt0() {
  asm volatile("s_wait_asynccnt 0" ::: "memory");
}

// ---------------------------------------------------------------------------
// Kernel 2: bilateral filter.
//   d2 = q + s(nb) - 2 * sum_c (rc*cen)[c]*nb[c]   (all in log2 domain)
//   q     : per output pixel, computed once
//   s(nb) : per tile pixel, precomputed cooperatively into LDS
// ---------------------------------------------------------------------------
template <int DYN>
__global__ __launch_bounds__(256) void bf_main(const float* __restrict__ in,
                                               const float* __restrict__ wsf,
                                               float*       __restrict__ out) {
  __shared__ __align__(16) float tile[LELEMS];   // [c][24][24]
  __shared__ float sq[LPIX];                     // per tile pixel: sum rc2*v^2

  const int tx = threadIdx.x, ty = threadIdx.y;
  const int tid = ty * TILE + tx;
  const int b  = blockIdx.z;
  const int y0 = blockIdx.y * TILE;
  const int x0 = blockIdx.x * TILE;

  const unsigned lbase = (unsigned)(uintptr_t)&tile[0];   // LDS byte offset

  // ---- cooperative halo-tile load: 1152 float4 chunks over 256 threads ----
  // Image dims are multiples of 4 and tile origins are multiples of 16, so
  // every 4-wide chunk is either fully inside or fully outside the image.
  #pragma unroll
  for (int i = 0; i < (LVEC4 + 255) / 256; ++i) {
    const int idx = i * 256 + tid;
    if (idx < LVEC4) {
      const int c   = idx / (LW * 6);            // 6 vec4 per 24-wide row
      const int rem = idx - c * (LW * 6);
      const int row = rem / 6;
      const int q4  = rem - row * 6;
      const int gy  = y0 - HALO + row;
      const int gx  = x0 - HALO + q4 * 4;
      const unsigned loff = lbase + (unsigned)idx * 16u;
      if ((unsigned)gy < (unsigned)H_ && (unsigned)gx < (unsigned)W_) {
        async_g2l_b128(loff, in + (((size_t)(b * C_ + c)) * H_ + gy) * W_ + gx);
      } else {
        *reinterpret_cast<float4*>(&tile[idx * 4]) = make_float4(0.f, 0.f, 0.f, 0.f);
      }
    }
  }

  // ---- uniform coefficients (scalar loads; overlap with async copies) ----
  float rc2[C_];
  #pragma unroll
  for (int c = 0; c < C_; ++c) rc2[c] = wsf[c];
  float cst[25];
  #pragma unroll
  for (int i = 0; i < 25; ++i) cst[i] = wsf[C_ + i];

  wait_asynccnt0();
  __syncthreads();

  // ---- precompute s(p) = sum_c rc2[c]*tile[c][p]^2 for all 576 tile pixels
  #pragma unroll
  for (int i = 0; i < (LPIX + 255) / 256; ++i) {
    const int p = i * 256 + tid;
    if (p < LPIX) {
      float acc = 0.f;
      #pragma unroll
      for (int c = 0; c < C_; ++c) {
        const float v = tile[c * LPIX + p];
        acc = fmaf(rc2[c] * v, v, acc);
      }
      sq[p] = acc;
    }
  }
  __syncthreads();

  // ---- per-pixel precompute: rcen[c] = rc2[c]*cen[c],  q = sum rcen*cen ----
  const int cy = ty + HALO, cx = tx + HALO;
  float rcen[C_];
  float q = 0.f;
  #pragma unroll
  for (int c = 0; c < C_; ++c) {
    const float v = tile[c * LPIX + cy * LW + cx];
    rcen[c] = rc2[c] * v;
    q = fmaf(rcen[c], v, q);
  }

  // ---- border validity masks (once, not per tap) ----
  bool vy[5], vx[5];
  #pragma unroll
  for (int k = 0; k < 5; ++k) {
    vy[k] = (unsigned)(y0 + ty + (k - 2) * 2) < (unsigned)H_;
    vx[k] = (unsigned)(x0 + tx + (k - 2) * 2) < (unsigned)W_;
  }

  float num[DYN];
  #pragma unroll
  for (int k = 0; k < DYN; ++k) num[k] = 0.f;
  float den = 0.f;

  #pragma unroll
  for (int ky = 0; ky < 5; ++ky) {
    #pragma unroll
    for (int kx = 0; kx < 5; ++kx) {
      const int off = (cy + (ky - 2) * 2) * LW + (cx + (kx - 2) * 2);
      const float* nbp = &tile[off];
      float nb[C_];
      float t = 0.f;
      #pragma unroll
      for (int c = 0; c < C_; ++c) {
        nb[c] = nbp[c * LPIX];
        t = fmaf(rcen[c], nb[c], t);
      }
      const float d2 = fmaf(t, -2.0f, q + sq[off]);
      const float cc = (vy[ky] && vx[kx]) ? cst[ky * 5 + kx] : -3.0e38f;
      const float w  = __builtin_amdgcn_exp2f(d2 + cc);   // exp2(-huge)==0
      den += w;
      #pragma unroll
      for (int k = 0; k < DYN; ++k) num[k] = fmaf(w, nb[k], num[k]);
    }
  }

  const int y = y0 + ty, x = x0 + tx;
  const float rden = 1.0f / den;                 // center tap guarantees den>0
  #pragma unroll
  for (int k = 0; k < DYN; ++k)
    out[(((size_t)(b * DYN + k)) * H_ + y) * W_ + x] = num[k] * rden;
}

// ---------------------------------------------------------------------------
extern "C" void kernel_launch(void* const* d_in, const int* in_sizes, int n_in,
                              void* d_out, int out_size, void* d_ws, size_t ws_size,
                              hipStream_t stream) {
  (void)in_sizes; (void)n_in; (void)ws_size;
  const float* inp  = (const float*)d_in[0];
  const float* prm  = (const float*)d_in[1];
  const float* kern = (const float*)d_in[2];
  const int*   dil  = (const int*)d_in[3];
  float* wsf = (float*)d_ws;
  float* out = (float*)d_out;

  bf_weight_xform<<<1, 32, 0, stream>>>(prm, kern, dil, wsf);

  const int dyn = out_size / (B_ * H_ * W_);     // dynamic_size
  dim3 grid(W_ / TILE, (H_ + TILE - 1) / TILE, B_);
  dim3 block(TILE, TILE);
  switch (dyn) {
    case 1: bf_main<1><<<grid, block, 0, stream>>>(inp, wsf, out); break;
    case 2: bf_main<2><<<grid, block, 0, stream>>>(inp, wsf, out); break;
    case 3: bf_main<3><<<grid, block, 0, stream>>>(inp, wsf, out); break;
    case 4: bf_main<4><<<grid, block, 0, stream>>>(inp, wsf, out); break;
    case 5: bf_main<5><<<grid, block, 0, stream>>>(inp, wsf, out); break;
    case 6: bf_main<6><<<grid, block, 0, stream>>>(inp, wsf, out); break;
    case 7: bf_main<7><<<grid, block, 0, stream>>>(inp, wsf, out); break;
    default: bf_main<8><<<grid, block, 0, stream>>>(inp, wsf, out); break;
  }
}